// L2Discrepancy_14139032338950
// MI455X (gfx1250) — compile-verified
//
#include <hip/hip_runtime.h>
#include <math.h>

typedef float v2f __attribute__((ext_vector_type(2)));
typedef float v8f __attribute__((ext_vector_type(8)));

#define L_LEN   2048
#define C_DIM   64
#define B_SZ    512
#define ROWS    128                 // seg rows per block
#define CHUNKS  (L_LEN / ROWS)      // 16 chunks per batch
#define STRIDE  66                  // padded LDS row stride in floats (8B aligned, conflict-free)

// ---- helpers --------------------------------------------------------------

// LDS byte-offset of a pointer into a __shared__ array (flat shared-aperture
// pointers carry the wave-relative LDS offset in the low 32 bits).
__device__ __forceinline__ unsigned lds_off(const void* p) {
    return (unsigned)(unsigned long long)p;
}

// Issue one async 8-byte global->LDS copy (ASYNCcnt-tracked, no VGPR round trip).
__device__ __forceinline__ void async_copy_b64(const float* gaddr, unsigned ldsoff) {
    asm volatile("global_load_async_to_lds_b64 %0, %1, off"
                 :: "v"(ldsoff), "v"(gaddr) : "memory");
}

__device__ __forceinline__ void wait_asynccnt0() {
    asm volatile("s_wait_asynccnt 0x0" ::: "memory");
}

// Stage A^T as float2 pairs: Ast[jp][i] = (A[i][2jp], A[i][2jp+1]) — B-operand layout.
__device__ __forceinline__ void stage_Ast(const float* __restrict__ Amat,
                                          float* Ast, int tid) {
    for (int idx = tid; idx < (C_DIM * C_DIM / 4); idx += 256) {
        const int flat = idx << 2;
        const int i    = flat >> 6;
        const int j    = flat & 63;        // multiple of 4
        const float4 a = *(const float4*)(Amat + i * C_DIM + j);
        *(v2f*)&Ast[((j >> 1) + 0) * 128 + i * 2] = (v2f){a.x, a.y};
        *(v2f*)&Ast[((j >> 1) + 1) * 128 + i * 2] = (v2f){a.z, a.w};
    }
}

// One wave computes a 16(rows) x 64(ch) tile of src @ A^T via chained
// V_WMMA_F32_16X16X4_F32. src rows are [m0, m0+16) of srcL.
__device__ __forceinline__ void wmma_tile(const float* srcL, const float* Ast,
                                          int m0, int lm, int h, v8f acc[4]) {
#pragma unroll
    for (int nt = 0; nt < 4; ++nt)
        acc[nt] = (v8f){0.f, 0.f, 0.f, 0.f, 0.f, 0.f, 0.f, 0.f};
#pragma unroll
    for (int kk = 0; kk < C_DIM; kk += 4) {
        // A operand (16x4 f32): lane half h holds K = kk+2h, kk+2h+1
        const v2f a = *(const v2f*)&srcL[(m0 + lm) * STRIDE + kk + 2 * h];
#pragma unroll
        for (int nt = 0; nt < 4; ++nt) {
            // B operand (4x16 f32): B[j][i] = A[i][j]
            const v2f bb = *(const v2f*)&Ast[((kk >> 1) + h) * 128 + (nt * 16 + lm) * 2];
            acc[nt] = __builtin_amdgcn_wmma_f32_16x16x4_f32(
                false, a, false, bb, (short)0, acc[nt], false, false);
        }
    }
}

// ---------------------------------------------------------------------------
// Kernel 0: y2[l][i] = sum_j A[i][j] * path2[l][j], for all 2048 rows.
// 16 blocks x 256 threads; async-stage path2, WMMA, store y2 to workspace.
// ---------------------------------------------------------------------------
__global__ __launch_bounds__(256)
void l2disc_y2(const float* __restrict__ path2,
               const float* __restrict__ Amat,
               float* __restrict__ y2)
{
    __shared__ float p2L[ROWS * STRIDE];
    __shared__ float Ast[32 * 128];

    const int tid = threadIdx.x;
    const int r0  = blockIdx.x * ROWS;

    for (int idx = tid; idx < ROWS * (C_DIM / 2); idx += 256) {
        const int row = idx >> 5;          // 32 pairs per row
        const int jp  = (idx & 31) << 1;   // even column
        async_copy_b64(path2 + (size_t)(r0 + row) * C_DIM + jp,
                       lds_off(&p2L[row * STRIDE + jp]));
    }
    stage_Ast(Amat, Ast, tid);
    wait_asynccnt0();
    __syncthreads();

    const int lane = tid & 31;
    const int wave = tid >> 5;     // wave32
    const int h    = lane >> 4;
    const int lm   = lane & 15;
    const int m0   = wave * 16;

    v8f acc[4];
    wmma_tile(p2L, Ast, m0, lm, h, acc);

#pragma unroll
    for (int nt = 0; nt < 4; ++nt) {
#pragma unroll
        for (int r = 0; r < 8; ++r)
            y2[(size_t)(r0 + m0 + r + 8 * h) * C_DIM + nt * 16 + lm] = acc[nt][r];
    }
}

// ---------------------------------------------------------------------------
// Main kernel: one block per (batch, 128-row chunk). Async-stage raw path1,
// y1 = tile @ A^T via WMMA, u = y1 - y2 (L2-resident), quadratic form + reduce.
// ---------------------------------------------------------------------------
__global__ __launch_bounds__(256)
void l2disc_main(const float* __restrict__ times,
                 const float* __restrict__ path1,
                 const float* __restrict__ y2,
                 const float* __restrict__ Amat,
                 float* __restrict__ partial)
{
    __shared__ float p1L[(ROWS + 1) * STRIDE]; // path1 tile, later overwritten with u = y1-y2
    __shared__ float Ast[32 * 128];
    __shared__ float red[256];

    const int tid   = threadIdx.x;
    const int b     = blockIdx.y;
    const int chunk = blockIdx.x;
    const int r0    = chunk * ROWS;

    // ---- async stage path1 rows r0 .. r0+128 (129 rows, +1 overlap) ----
    const float* p1 = path1 + (size_t)b * L_LEN * C_DIM;
    for (int idx = tid; idx < (ROWS + 1) * (C_DIM / 2); idx += 256) {
        const int row  = idx >> 5;
        const int jp   = (idx & 31) << 1;
        const int grow = r0 + row;
        if (grow < L_LEN)
            async_copy_b64(p1 + (size_t)grow * C_DIM + jp,
                           lds_off(&p1L[row * STRIDE + jp]));
    }
    // zero the OOB overlap row (last chunk only)
    if (r0 + ROWS >= L_LEN && tid < 32)
        *(v2f*)&p1L[ROWS * STRIDE + tid * 2] = (v2f){0.f, 0.f};

    stage_Ast(Amat, Ast, tid);
    wait_asynccnt0();
    __syncthreads();

    // ---- WMMA phase: wave w owns rows [w*16, w*16+16) ----
    const int lane = tid & 31;
    const int wave = tid >> 5;
    const int h    = lane >> 4;
    const int lm   = lane & 15;
    const int m0   = wave * 16;

    v8f acc[4];
    wmma_tile(p1L, Ast, m0, lm, h, acc);

    // ---- overlap row 128: y1 via VALU dot, minus y2 (read before overwrite) ----
    float ub = 0.f;
    if (tid < C_DIM) {
#pragma unroll 8
        for (int j = 0; j < C_DIM; ++j)
            ub = fmaf(Amat[tid * C_DIM + j], p1L[ROWS * STRIDE + j], ub);
        const int grow = r0 + ROWS;
        if (grow < L_LEN) ub -= y2[(size_t)grow * C_DIM + tid];
    }

    // ---- write u = y1 - y2 over the tile (each wave writes only its own rows) ----
#pragma unroll
    for (int nt = 0; nt < 4; ++nt) {
#pragma unroll
        for (int r = 0; r < 8; ++r) {
            const int row = m0 + r + 8 * h;
            const int col = nt * 16 + lm;
            p1L[row * STRIDE + col] =
                acc[nt][r] - y2[(size_t)(r0 + row) * C_DIM + col];
        }
    }
    if (tid < C_DIM) p1L[ROWS * STRIDE + tid] = ub;
    __syncthreads();

    // ---- per-row quadratic form: seg[l] = sum_i u^2 + u v + v^2, * dt ----
    float val = 0.f;
    if (tid < ROWS) {
        const int l = r0 + tid;
        if (l < L_LEN - 1) {
            const float* ur = &p1L[tid * STRIDE];
            const float* vr = &p1L[(tid + 1) * STRIDE];
            float s = 0.f;
#pragma unroll 8
            for (int i = 0; i < C_DIM; ++i) {
                const float u = ur[i];
                const float v = vr[i];
                s = fmaf(u, u, s);
                s = fmaf(u, v, s);
                s = fmaf(v, v, s);
            }
            val = s * (times[l + 1] - times[l]);
        }
    }
    red[tid] = val;
    __syncthreads();
#pragma unroll
    for (int off = 128; off > 0; off >>= 1) {
        if (tid < off) red[tid] += red[tid + off];
        __syncthreads();
    }
    if (tid == 0) partial[b * CHUNKS + chunk] = red[0];
}

// ---------------------------------------------------------------------------
// Final: per-batch deterministic sum of 16 partials, sqrt(s/3).
// ---------------------------------------------------------------------------
__global__ __launch_bounds__(256)
void l2disc_final(const float* __restrict__ partial, float* __restrict__ out)
{
    const int g = blockIdx.x * blockDim.x + threadIdx.x;
    if (g < B_SZ) {
        float s = 0.f;
#pragma unroll
        for (int c = 0; c < CHUNKS; ++c) s += partial[g * CHUNKS + c];
        out[g] = sqrtf(s / 3.0f);
    }
}

extern "C" void kernel_launch(void* const* d_in, const int* in_sizes, int n_in,
                              void* d_out, int out_size, void* d_ws, size_t ws_size,
                              hipStream_t stream)
{
    const float* times = (const float*)d_in[0];
    const float* path1 = (const float*)d_in[1];
    const float* path2 = (const float*)d_in[2];
    const float* Amat  = (const float*)d_in[3];
    float* out     = (float*)d_out;
    float* partial = (float*)d_ws;                       // B_SZ*CHUNKS floats = 32 KB
    float* y2      = (float*)d_ws + B_SZ * CHUNKS;       // L_LEN*C_DIM floats = 512 KB

    l2disc_y2<<<CHUNKS, 256, 0, stream>>>(path2, Amat, y2);
    dim3 grid(CHUNKS, B_SZ, 1);
    l2disc_main<<<grid, 256, 0, stream>>>(times, path1, y2, Amat, partial);
    l2disc_final<<<(B_SZ + 255) / 256, 256, 0, stream>>>(partial, out);
}